// MC2ColumnParallelLinearImpl_44856638439896
// MI455X (gfx1250) — compile-verified
//
#include <hip/hip_runtime.h>

typedef __attribute__((ext_vector_type(2))) float        v2f;
typedef __attribute__((ext_vector_type(8))) float        v8f;
typedef __attribute__((ext_vector_type(4))) unsigned int v4u;
typedef __attribute__((ext_vector_type(8))) int          v8i;
typedef __attribute__((ext_vector_type(4))) int          v4i;

#define TM   128   // M tile per workgroup
#define TN    64   // N tile per workgroup
#define KB    32   // K block staged in LDS (floats)
#define LDA   36   // padded LDS row stride in floats (32 + 4 pad, produced by TDM pad)

// LDS arena layout (floats). Static LDS allocation starts at byte offset 0.
#define A_TILE_FLTS (TM * LDA)                 // 4608
#define W_TILE_FLTS (TN * LDA)                 // 2304
#define OFF_A(buf)  ((buf) * A_TILE_FLTS)
#define OFF_W(buf)  (2 * A_TILE_FLTS + (buf) * W_TILE_FLTS)
#define LDS_TOTAL   (2 * A_TILE_FLTS + 2 * W_TILE_FLTS)   // 13824 floats = 54 KB

// Issue one TDM 2D tile load: tile_d1 rows x tile_d0 elems (fp32) from gptr
// (row stride = stride0 elems) into LDS at lds_byte_off, padding the LDS
// destination by 4 DWORDs after every 32 DWORDs (-> 36-float row stride).
__device__ __forceinline__ void tdm_load_tile_2d(
    unsigned lds_byte_off, const float* gptr,
    unsigned tensor_d0, unsigned tensor_d1,
    unsigned tile_d0, unsigned tile_d1, unsigned stride0)
{
    const unsigned long long ga = (unsigned long long)(uintptr_t)gptr;

    v4u g0;
    g0[0] = 1u;                                         // count=1, user desc, no gather
    g0[1] = lds_byte_off;                               // lds_addr [63:32]
    g0[2] = (unsigned)(ga & 0xFFFFFFFFu);               // global_addr [95:64]
    g0[3] = (unsigned)((ga >> 32) & 0x01FFFFFFu)        // global_addr [120:96]
          | (2u << 30);                                 // type=2 ("image") [127:126]

    v8i g1;
    g1[0] = (int)((2u << 16)        // data_size = 2 -> 4 bytes
                | (1u << 20)        // pad_enable
                | (4u << 22)        // pad_interval code 4 -> every 32 DWORDs
                | (3u << 25));      // pad_amount  code 3 -> 4 DWORDs
    g1[1] = (int)((tensor_d0 & 0xFFFFu) << 16);                    // dim0[15:0]; abar=0
    g1[2] = (int)((tensor_d0 >> 16) | ((tensor_d1 & 0xFFFFu) << 16));
    g1[3] = (int)((tensor_d1 >> 16) | (tile_d0 << 16));            // tile_dim0
    g1[4] = (int)tile_d1;                                          // tile_dim1; tile_dim2=0
    g1[5] = (int)stride0;                                          // dim0_stride[31:0]
    g1[6] = 0;                                                     // dim0_stride[47:32]; dim1_stride lo
    g1[7] = 0;

    v4i g2 = {0, 0, 0, 0};          // 2D tile: higher-dim groups unused
    v4i g3 = {0, 0, 0, 0};
    v8i g4 = {0, 0, 0, 0, 0, 0, 0, 0};

    // clang-23 / therock-10.0 lane: 6-arg form (v4u, v8i, v4i, v4i, v8i, i32 cpol)
    __builtin_amdgcn_tensor_load_to_lds(g0, g1, g2, g3, g4, 0);
}

__global__ __launch_bounds__(256) void colparallel_linear_wmma_tdm(
    const float* __restrict__ x,     // [M, K]
    const float* __restrict__ w,     // [N, K]
    const float* __restrict__ bias,  // [N]
    float* __restrict__ out,         // [M, N]
    int Mtot, int Kdim, int Ndim)
{
    __shared__ float lds[LDS_TOTAL];

    // Escape the LDS arena: ptrtoint + asm capture marks it as externally
    // written, so the TDM's (unmodeled) LDS writes can alias it and the
    // ds_loads below cannot be folded to poison.
    {
        unsigned long long esc = (unsigned long long)(uintptr_t)&lds[0];
        asm volatile("" : "+v"(esc) :: "memory");
    }

    const int tid  = threadIdx.x;
    const int lane = tid & 31;
    // Scalar wave id (readfirstlane -> SGPR) so the TDM-issue branch is a
    // true uniform s_cbranch, not an exec-mask dance.
    const int waveS = __builtin_amdgcn_readfirstlane(tid) >> 5;
    const int wave  = tid >> 5;

    const int tileM = blockIdx.y * TM;
    const int tileN = blockIdx.x * TN;

    const float* xTile = x + (size_t)tileM * Kdim;   // row tileM, col 0
    const float* wTile = w + (size_t)tileN * Kdim;

    v8f acc0 = {}, acc1 = {}, acc2 = {}, acc3 = {};

    // WMMA f32 16x16x4 operand addressing (ISA VGPR layout):
    // lanes 0-15 hold K={0,1}, lanes 16-31 hold K={2,3}; row/col = lane&15.
    const int rowm  = lane & 15;
    const int khalf = (lane >> 4) << 1;   // 0 or 2

    const int NBLK = Kdim / KB;

    // ---- prologue: TDM-load block 0 into buffer 0 ----
    if (waveS == 0) {
        tdm_load_tile_2d(OFF_A(0) * 4u, xTile, Kdim, Mtot, KB, TM, Kdim);
        tdm_load_tile_2d(OFF_W(0) * 4u, wTile, Kdim, Ndim, KB, TN, Kdim);
    }

    for (int blk = 0; blk < NBLK; ++blk) {
        const int buf = blk & 1;

        if (waveS == 0) {
            if (blk + 1 < NBLK) {
                // Prefetch next K block into the other buffer (its consumers
                // finished at the trailing barrier of iteration blk-1).
                const int kb = (blk + 1) * KB;
                tdm_load_tile_2d(OFF_A(buf ^ 1) * 4u, xTile + kb, Kdim, Mtot, KB, TM, Kdim);
                tdm_load_tile_2d(OFF_W(buf ^ 1) * 4u, wTile + kb, Kdim, Ndim, KB, TN, Kdim);
                // In-order TENSORcnt: <=2 outstanding => block `blk`'s 2 ops done.
                __builtin_amdgcn_s_wait_tensorcnt(2);
            } else {
                __builtin_amdgcn_s_wait_tensorcnt(0);
            }
        }
        __syncthreads();               // release all waves onto buffer `buf`
        asm volatile("" ::: "memory"); // pin LDS loads inside this iteration

        const float* As = &lds[OFF_A(buf)];
        const float* Ws = &lds[OFF_W(buf)];

        #pragma unroll
        for (int ks = 0; ks < KB; ks += 4) {
            const int kk = ks + khalf;
            const v2f a  = *(const v2f*)&As[(wave * 16 + rowm) * LDA + kk];
            const v2f b0 = *(const v2f*)&Ws[( 0 + rowm) * LDA + kk];
            const v2f b1 = *(const v2f*)&Ws[(16 + rowm) * LDA + kk];
            const v2f b2 = *(const v2f*)&Ws[(32 + rowm) * LDA + kk];
            const v2f b3 = *(const v2f*)&Ws[(48 + rowm) * LDA + kk];
            acc0 = __builtin_amdgcn_wmma_f32_16x16x4_f32(
                false, a, false, b0, (short)0, acc0, false, false);
            acc1 = __builtin_amdgcn_wmma_f32_16x16x4_f32(
                false, a, false, b1, (short)0, acc1, false, false);
            acc2 = __builtin_amdgcn_wmma_f32_16x16x4_f32(
                false, a, false, b2, (short)0, acc2, false, false);
            acc3 = __builtin_amdgcn_wmma_f32_16x16x4_f32(
                false, a, false, b3, (short)0, acc3, false, false);
        }
        __syncthreads();   // buffer `buf` free for TDM overwrite next iteration
    }

    // ---- epilogue: bias add + store ----
    // C/D layout: VGPR v -> M = v (lanes 0-15) or v+8 (lanes 16-31); N = lane&15.
    const int col    = lane & 15;
    const int mhi    = (lane >> 4) << 3;
    const int gmBase = tileM + wave * 16 + mhi;

    v8f accs[4] = {acc0, acc1, acc2, acc3};
    #pragma unroll
    for (int n = 0; n < 4; ++n) {
        const int   gn = tileN + n * 16 + col;
        const float bv = bias[gn];
        #pragma unroll
        for (int v = 0; v < 8; ++v) {
            out[(size_t)(gmBase + v) * Ndim + gn] = accs[n][v] + bv;
        }
    }
}

extern "C" void kernel_launch(void* const* d_in, const int* in_sizes, int n_in,
                              void* d_out, int out_size, void* d_ws, size_t ws_size,
                              hipStream_t stream) {
    const float* x    = (const float*)d_in[0];  // [8192, 4, 1024] -> [32768, 1024]
    const float* w    = (const float*)d_in[1];  // [4096, 1024]
    const float* bias = (const float*)d_in[2];  // [4096]
    float* out        = (float*)d_out;          // [32768, 4096]

    const int Kdim = 1024;
    const int Ndim = 4096;
    const int Mtot = in_sizes[0] / Kdim;        // 32768

    dim3 grid(Ndim / TN, Mtot / TM);            // (64, 256); x fastest keeps X-tile L2-hot
    dim3 block(256);
    hipLaunchKernelGGL(colparallel_linear_wmma_tdm, grid, block, 0, stream,
                       x, w, bias, out, Mtot, Kdim, Ndim);
}